// EdgePredictor_30459908063742
// MI455X (gfx1250) — compile-verified
//
#include <hip/hip_runtime.h>
#include <hip/hip_bf16.h>
#include <math.h>

typedef __attribute__((ext_vector_type(2))) float v2f;
typedef __attribute__((ext_vector_type(8))) float v8f;

#define WMMA4(a, b, c) \
    __builtin_amdgcn_wmma_f32_16x16x4_f32(false, (a), false, (b), (short)0, (c), false, false)

namespace {
constexpr int kB = 4;
constexpr int kN = 1024;
constexpr int kD = 3;
constexpr int kH = 128;           // layer1 width
constexpr int kH2 = 64;           // layer2 width
constexpr int kE = kN * (kN - 1) / 2;      // 523776 edges
constexpr int kRows = kB * kE;             // 2095104 rows
constexpr int kWavesPerBlock = 8;
constexpr int kThreads = kWavesPerBlock * 32;
constexpr int kRowsPerBlock = kWavesPerBlock * 16;  // 128
constexpr int kBlocks = kRows / kRowsPerBlock;      // 16368 (exact)

// dynamic LDS layout (floats)
constexpr int kOffW1 = 0;                    // [8][128] zero-padded rows 6,7
constexpr int kOffW2 = kOffW1 + 8 * kH;      // [128][64]
constexpr int kOffB1 = kOffW2 + kH * kH2;    // [128]
constexpr int kOffB2 = kOffB1 + kH;          // [64]
constexpr int kOffW3 = kOffB2 + kH2;         // [64]
constexpr int kOffH1 = kOffW3 + kH2;         // per-wave [16][128]
constexpr int kH1PerWave = 16 * kH;          // 2048 floats
constexpr int kLdsFloats = kOffH1 + kWavesPerBlock * kH1PerWave;
constexpr size_t kLdsBytes = (size_t)kLdsFloats * sizeof(float);  // 103424 B

__device__ __forceinline__ int triStart(int i) {
    // number of edges before row i in triu(k=1) order
    return i * (2 * kN - 1 - i) / 2;
}
}  // namespace

__global__ __launch_bounds__(kThreads)
void EdgePredictor_30459908063742_kernel(const float* __restrict__ V,
                                         const float* __restrict__ gW1,
                                         const float* __restrict__ gb1,
                                         const float* __restrict__ gW2,
                                         const float* __restrict__ gb2,
                                         const float* __restrict__ gW3,
                                         const float* __restrict__ gb3,
                                         float* __restrict__ out) {
    extern __shared__ float smem[];
    float* sW1 = smem + kOffW1;
    float* sW2 = smem + kOffW2;
    float* sB1 = smem + kOffB1;
    float* sB2 = smem + kOffB2;
    float* sW3 = smem + kOffW3;

    const int tid = threadIdx.x;
    // ---- stage weights into LDS (shared by all 8 waves) ----
    for (int idx = tid; idx < 8 * kH; idx += kThreads) {
        int k = idx >> 7, n = idx & 127;
        sW1[idx] = (k < 2 * kD) ? gW1[k * kH + n] : 0.0f;  // zero-pad K=6,7
    }
    for (int idx = tid; idx < kH * kH2; idx += kThreads) sW2[idx] = gW2[idx];
    if (tid < kH) sB1[tid] = gb1[tid];
    if (tid < kH2) { sB2[tid] = gb2[tid]; sW3[tid] = gW3[tid]; }
    __syncthreads();

    const int wave = tid >> 5;
    const int lane = tid & 31;
    const int half = lane >> 4;   // 0: K pair {0,1}; 1: K pair {2,3} of each k-step
    const int l15  = lane & 15;
    float* sH1 = smem + kOffH1 + wave * kH1PerWave;  // per-wave [16][128]

    const int tile = blockIdx.x * kWavesPerBlock + wave;
    const int r0 = tile * 16;              // first global row of this tile
    const int b  = r0 / kE;                // batch (tiles never straddle batches)
    const int e0 = r0 - b * kE;            // first edge index

    // ---- decode (i, j) for this lane's row m = l15 ----
    const int em = e0 + l15;
    double disc = (double)(2 * kN - 1) * (2 * kN - 1) - 8.0 * (double)em;
    int i = (int)(((double)(2 * kN - 1) - sqrt(disc)) * 0.5);
    if (i < 0) i = 0;
    if (i > kN - 2) i = kN - 2;
    while (i < kN - 2 && triStart(i + 1) <= em) ++i;
    while (i > 0 && triStart(i) > em) --i;
    const int j = i + 1 + (em - triStart(i));

    // ---- gather edge features: concat(V[b,i,:], V[b,j,:]) ----
    const float* vi = V + ((size_t)b * kN + i) * kD;
    const float* vj = V + ((size_t)b * kN + j) * kD;
    const float f0 = vi[0], f1 = vi[1], f2 = vi[2];
    const float f3 = vj[0], f4 = vj[1], f5 = vj[2];

    // A fragments for layer 1 (16x4 fp32 A layout: lanes 0-15 hold K={0,1}, 16-31 K={2,3})
    v2f a1s0, a1s1;
    a1s0.x = half ? f2 : f0;  a1s0.y = half ? f3 : f1;      // k-step 0: K 0..3
    a1s1.x = half ? 0.f : f4; a1s1.y = half ? 0.f : f5;     // k-step 1: K 4..7 (6,7 pad)

    // ---- layer 1: h1[16,128] = relu(feat @ W1 + b1), staged to LDS ----
#pragma unroll
    for (int t = 0; t < 8; ++t) {
        const int n = 16 * t + l15;
        v2f b0, b1v;
        b0.x  = sW1[(2 * half + 0) * kH + n];
        b0.y  = sW1[(2 * half + 1) * kH + n];
        b1v.x = sW1[(4 + 2 * half + 0) * kH + n];
        b1v.y = sW1[(4 + 2 * half + 1) * kH + n];
        v8f c = {};
        c = WMMA4(a1s0, b0, c);
        c = WMMA4(a1s1, b1v, c);
        const float bias = sB1[n];
#pragma unroll
        for (int r = 0; r < 8; ++r) {
            float v = c[r] + bias;
            v = fmaxf(v, 0.0f);
            sH1[(r + 8 * half) * kH + n] = v;   // D layout -> row-major h1 in LDS
        }
    }

    // ---- layer 2: h2[16,64] = relu(h1 @ W2 + b2), accumulated in 4 C tiles ----
    v8f c2[4];
#pragma unroll
    for (int t = 0; t < 4; ++t) c2[t] = (v8f){};

    for (int s = 0; s < 32; ++s) {                 // K = 128 in steps of 4
        const int k = 4 * s + 2 * half;
        const float2 hv = *(const float2*)&sH1[l15 * kH + k];
        v2f a2; a2.x = hv.x; a2.y = hv.y;
#pragma unroll
        for (int t = 0; t < 4; ++t) {
            const int n = 16 * t + l15;
            v2f bb;
            bb.x = sW2[(k + 0) * kH2 + n];
            bb.y = sW2[(k + 1) * kH2 + n];
            c2[t] = WMMA4(a2, bb, c2[t]);
        }
    }

    // bias + relu in C layout (each lane owns a fixed column per tile)
    float w3v[4];
#pragma unroll
    for (int t = 0; t < 4; ++t) {
        const int n = 16 * t + l15;
        const float bias = sB2[n];
        w3v[t] = sW3[n];
#pragma unroll
        for (int r = 0; r < 8; ++r) c2[t][r] = fmaxf(c2[t][r] + bias, 0.0f);
    }

    // ---- layer 3: sigmoid(h2 @ W3 + b3) via 16-lane shuffle reduction ----
    const float b3v = gb3[0];
#pragma unroll
    for (int r = 0; r < 8; ++r) {
        float p = c2[0][r] * w3v[0] + c2[1][r] * w3v[1] +
                  c2[2][r] * w3v[2] + c2[3][r] * w3v[3];
        p += __shfl_xor(p, 1);
        p += __shfl_xor(p, 2);
        p += __shfl_xor(p, 4);
        p += __shfl_xor(p, 8);      // full sum within each 16-lane group
        if (l15 == r) {
            const int m = r + 8 * half;              // row within tile for this group
            const float logit = p + b3v;
            const float prob = 1.0f / (1.0f + __expf(-logit));
            out[(size_t)b * kE + e0 + m] = prob;
        }
    }

    // ---- edge_indices output (written once, by batch-0 tiles), as floats ----
    if (b == 0 && lane < 16) {
        float* oi = out + (size_t)kRows + 2 * (size_t)em;
        oi[0] = (float)i;
        oi[1] = (float)j;
    }
}

extern "C" void kernel_launch(void* const* d_in, const int* in_sizes, int n_in,
                              void* d_out, int out_size, void* d_ws, size_t ws_size,
                              hipStream_t stream) {
    (void)in_sizes; (void)n_in; (void)d_ws; (void)ws_size; (void)out_size;
    const float* V  = (const float*)d_in[0];
    const float* W1 = (const float*)d_in[1];
    const float* b1 = (const float*)d_in[2];
    const float* W2 = (const float*)d_in[3];
    const float* b2 = (const float*)d_in[4];
    const float* W3 = (const float*)d_in[5];
    const float* b3 = (const float*)d_in[6];
    float* out = (float*)d_out;

    EdgePredictor_30459908063742_kernel<<<kBlocks, kThreads, kLdsBytes, stream>>>(
        V, W1, b1, W2, b2, W3, b3, out);
}